// Opacoxel_15032385536488
// MI455X (gfx1250) — compile-verified
//
#include <hip/hip_runtime.h>

// Trilinear interpolation of a 256^3 fp32 grid at N points + sigmoid.
// MI455X (gfx1250) design:
//  - 64MB grid stays L2-resident (192MB L2); all streaming traffic
//    (positions in, results out) uses non-temporal hints.
//  - Positions are software-pipelined through a wave-private LDS double
//    buffer with GLOBAL_LOAD_ASYNC_TO_LDS_B32 (ASYNCcnt, GVS/saddr form),
//    hiding streaming-load latency behind the gather+lerp of the previous
//    chunk. Wave-private buffers => no barriers needed (wave32).
//  - 8 corners gathered as 4 z-contiguous b64 pairs, 7 FMA lerps.
//  - sigmoid via v_exp_f32 (2^x) + v_rcp_f32.
//  - All index math is 32-bit (N <= 4.2M, buffers <= 64MB).

#define GN   256
#define GSX  (GN * GN)   // x stride in elements
#define GSY  (GN)        // y stride in elements

__device__ __forceinline__ int clampi(int v, int lo, int hi) {
    return v < lo ? lo : (v > hi ? hi : v);
}

// Stage one wave-chunk (32 points = 96 dwords = 384B) of positions into LDS.
// Three wave-wide async b32 copies, saddr-form: uniform base in SGPRs,
// per-lane 32-bit byte offset (clamped in-bounds for the ragged tail).
__device__ __forceinline__ void async_pos_chunk(const float* pos, int dwTotal,
                                                int ptBase, unsigned ldsDst,
                                                int lane)
{
    const int dwBase = 3 * ptBase;
    const unsigned long long gbase = (unsigned long long)(const void*)pos;
#pragma unroll
    for (int j = 0; j < 3; ++j) {
        int dw = dwBase + lane + j * 32;
        if (dw > dwTotal - 1) dw = dwTotal - 1;          // clamp ragged tail
        unsigned voff = 4u * (unsigned)dw;               // byte offset < 2^31
        unsigned ldst = ldsDst + 4u * (unsigned)(lane + j * 32);
        // memory -> LDS, tracked by ASYNCcnt; NT so the stream doesn't
        // evict the L2-resident grid.
        asm volatile("global_load_async_to_lds_b32 %0, %1, %2 th:TH_LOAD_NT"
                     :: "v"(ldst), "v"(voff), "s"(gbase) : "memory");
    }
}

__global__ __launch_bounds__(256) void opacoxel_trilerp_sigmoid(
    const float* __restrict__ pos,    // (N,3) row-major
    const float* __restrict__ grid,   // (256,256,256) C-order
    float* __restrict__ out,          // (N,1)
    int n)
{
    __shared__ float sbuf[8][2][96];  // 8 waves x double buffer x 384B = 6KB

    const int lane = (int)(threadIdx.x & 31u);
    const int wid  = (int)(threadIdx.x >> 5);
    const int wavesPerBlock = (int)(blockDim.x >> 5);
    const int gwave = (int)blockIdx.x * wavesPerBlock + wid;
    const int step  = (int)gridDim.x * wavesPerBlock * 32;
    const int dwTotal = 3 * n;

    const int base = gwave * 32;
    if (base >= n) return;

    const unsigned ldsA = (unsigned)(size_t)(void*)&sbuf[wid][0][0];
    const unsigned ldsB = (unsigned)(size_t)(void*)&sbuf[wid][1][0];

    // Prologue: stage chunk 0.                         outstanding async: 3
    async_pos_chunk(pos, dwTotal, base, ldsA, lane);

    int buf = 0;
    for (int c = base; c < n; c += step) {
        // Stage the NEXT chunk into the other buffer (dummy re-copy of the
        // current chunk if there is no next, keeping the counter protocol
        // uniform).                                    outstanding: <=6
        int nxt = c + step;
        async_pos_chunk(pos, dwTotal, (nxt < n && nxt >= 0) ? nxt : c,
                        buf ? ldsA : ldsB, lane);

        // 3 copies just issued; ASYNCcnt <= 3 ==> the 3 copies for the
        // chunk we read below have completed (in-order completion).
        asm volatile("s_wait_asynccnt 0x3" ::: "memory");

        const float* s = buf ? &sbuf[wid][1][0] : &sbuf[wid][0][0];
        float px = s[3 * lane + 0];
        float py = s[3 * lane + 1];
        float pz = s[3 * lane + 2];

        // bounds (-1,1) per axis, 256 cells: gc = (p+1)/2 * 255
        float gx = (px + 1.0f) * 127.5f;
        float gy = (py + 1.0f) * 127.5f;
        float gz = (pz + 1.0f) * 127.5f;

        int ix = clampi((int)floorf(gx), 0, GN - 2);
        int iy = clampi((int)floorf(gy), 0, GN - 2);
        int iz = clampi((int)floorf(gz), 0, GN - 2);

        // Fraction relative to the clamped cell; clamping it to [0,1]
        // reproduces the reference's index-clamp edge behavior.
        float xd = fminf(fmaxf(gx - (float)ix, 0.0f), 1.0f);
        float yd = fminf(fmaxf(gy - (float)iy, 0.0f), 1.0f);
        float zd = fminf(fmaxf(gz - (float)iz, 0.0f), 1.0f);

        // Gather 8 corners as 4 z-contiguous pairs (L2-resident grid).
        const float* g00 = grid + ((unsigned)ix * GSX + (unsigned)iy * GSY + (unsigned)iz);
        const float* g01 = g00 + GSY;   // (x0, y1)
        const float* g10 = g00 + GSX;   // (x1, y0)
        const float* g11 = g10 + GSY;   // (x1, y1)

        float c000 = g00[0], c001 = g00[1];
        float c010 = g01[0], c011 = g01[1];
        float c100 = g10[0], c101 = g10[1];
        float c110 = g11[0], c111 = g11[1];

        // Lerp z (contiguous axis) first, then y, then x: 7 FMAs.
        float c00 = fmaf(zd, c001 - c000, c000);
        float c01 = fmaf(zd, c011 - c010, c010);
        float c10 = fmaf(zd, c101 - c100, c100);
        float c11 = fmaf(zd, c111 - c110, c110);
        float c0  = fmaf(yd, c01 - c00, c00);
        float c1  = fmaf(yd, c11 - c10, c10);
        float v   = fmaf(xd, c1 - c0, c0);

        // sigmoid(v) = 1 / (1 + exp(-v)) via v_exp_f32 + v_rcp_f32.
        float e = __builtin_amdgcn_exp2f(v * -1.44269504088896340736f);
        float sv = __builtin_amdgcn_rcpf(1.0f + e);

        int i = c + lane;
        if (i < n) __builtin_nontemporal_store(sv, out + i);

        buf ^= 1;
    }
    // Up to 3 dummy async copies may remain outstanding; S_ENDPGM performs
    // an implicit wait-idle, so they drain before wave termination.
}

extern "C" void kernel_launch(void* const* d_in, const int* in_sizes, int n_in,
                              void* d_out, int out_size, void* d_ws, size_t ws_size,
                              hipStream_t stream) {
    const float* pos  = (const float*)d_in[0];   // (N,3) fp32
    const float* grid = (const float*)d_in[1];   // 256^3 fp32
    float* out = (float*)d_out;                  // (N,1) fp32

    int n = in_sizes[0] / 3;
    const int block = 256;                 // 8 wave32s per block
    int blocks = (n + block - 1) / block;
    if (blocks > 4096) blocks = 4096;      // grid-stride: ~1M lanes, 4 chunks/wave
    if (blocks < 1) blocks = 1;

    hipLaunchKernelGGL(opacoxel_trilerp_sigmoid,
                       dim3(blocks), dim3(block), 0, stream,
                       pos, grid, out, n);
}